// ChamferDist_12128987644324
// MI455X (gfx1250) — compile-verified
//
#include <hip/hip_runtime.h>
#include <hip/hip_bf16.h>

typedef float v2f __attribute__((ext_vector_type(2)));
typedef float v8f __attribute__((ext_vector_type(8)));

#define EPSV      1e-08f
#define DIST_MAXV 1.0e9f
#define CHUNK     2048     // target points staged in LDS per pass (24 KB)

// One-direction chamfer: for each query point, min/argmin of squared distance
// over all target points.  D = A(16x4) x B(4x16) + C  via v_wmma_f32_16x16x4_f32.
//   A[q,:] = (-2qx, -2qy, -2qz, 1)     B[:,t] = (tx, ty, tz, |t|^2 + EPS)
//   C[r]   = |q_row|^2  (loop-invariant, row-indexed -> no per-iter C movs)
// => D[q,t] = |q|^2 - 2 q.t + |t|^2 + EPS  (clamp applied after the min).
// Each wave owns TWO 16-query tiles sharing one B operand per iteration, so
// the per-iteration B/t2e setup amortizes over two WMMAs.
__global__ __launch_bounds__(256) void chamfer_dir_kernel(
    const float* __restrict__ qpts,   // [B, nQ, 3]
    const float* __restrict__ tpts,   // [B, nT, 3]
    float* __restrict__ outMin,       // [B, nQ]
    float* __restrict__ outIdx,       // [B, nQ]  (argmin written as float)
    int nQ, int nT)
{
    __shared__ float sT[CHUNK * 3];

    const int lane = threadIdx.x & 31;
    const int wave = threadIdx.x >> 5;
    const int b    = blockIdx.y;
    const int l16  = lane & 15;
    const bool hi  = lane >= 16;

    const float* __restrict__ qb = qpts + (size_t)b * nQ * 3;
    const float* __restrict__ tb = tpts + (size_t)b * nT * 3;

    // ---- A operands: two 16-query tiles per wave (16x4 f32 A layout) ----
    const int qbase = (blockIdx.x * 8 + wave) * 32;

    v2f amat0, amat1;
    v8f cinit0, cinit1;
    {
        const int qi0 = qbase + l16;
        const float x0 = qb[qi0 * 3 + 0];
        const float y0 = qb[qi0 * 3 + 1];
        const float z0 = qb[qi0 * 3 + 2];
        const float q2_0 = x0 * x0 + y0 * y0 + z0 * z0;
        amat0.x = hi ? (-2.0f * z0) : (-2.0f * x0);   // K=2 : K=0
        amat0.y = hi ? 1.0f         : (-2.0f * y0);   // K=3 : K=1

        const int qi1 = qbase + 16 + l16;
        const float x1 = qb[qi1 * 3 + 0];
        const float y1 = qb[qi1 * 3 + 1];
        const float z1 = qb[qi1 * 3 + 2];
        const float q2_1 = x1 * x1 + y1 * y1 + z1 * z1;
        amat1.x = hi ? (-2.0f * z1) : (-2.0f * x1);
        amat1.y = hi ? 1.0f         : (-2.0f * y1);

        // loop-invariant C: |q|^2 of row r (+8 for the hi lane half)
#pragma unroll
        for (int r = 0; r < 8; ++r) {
            cinit0[r] = __shfl(q2_0, r + (hi ? 8 : 0), 32);
            cinit1[r] = __shfl(q2_1, r + (hi ? 8 : 0), 32);
        }
    }

    float rmin0[8], rmin1[8];
    int   ridx0[8], ridx1[8];
#pragma unroll
    for (int r = 0; r < 8; ++r) {
        rmin0[r] = 3.0e38f; ridx0[r] = 0;
        rmin1[r] = 3.0e38f; ridx1[r] = 0;
    }

    for (int cbase = 0; cbase < nT; cbase += CHUNK) {
        __syncthreads();
        // cooperative stage of CHUNK target points into LDS
        for (int i = threadIdx.x; i < CHUNK * 3; i += 256)
            sT[i] = tb[(size_t)cbase * 3 + i];
        __syncthreads();

#pragma unroll 2
        for (int tt = 0; tt < CHUNK; tt += 16) {
            const int tl = (tt + l16) * 3;
            const float tx = sT[tl + 0];
            const float ty = sT[tl + 1];
            const float tz = sT[tl + 2];
            const float t2e = tx * tx + ty * ty + tz * tz + EPSV;

            v2f bmat;
            bmat.x = hi ? tz  : tx;   // K=2 : K=0
            bmat.y = hi ? t2e : ty;   // K=3 : K=1

            const v8f d0 = __builtin_amdgcn_wmma_f32_16x16x4_f32(
                    false, amat0, false, bmat, (short)0, cinit0, false, false);
            const v8f d1 = __builtin_amdgcn_wmma_f32_16x16x4_f32(
                    false, amat1, false, bmat, (short)0, cinit1, false, false);

            const int tglob = cbase + tt + l16;
#pragma unroll
            for (int r = 0; r < 8; ++r) {
                if (d0[r] < rmin0[r]) { rmin0[r] = d0[r]; ridx0[r] = tglob; }
            }
#pragma unroll
            for (int r = 0; r < 8; ++r) {
                if (d1[r] < rmin1[r]) { rmin1[r] = d1[r]; ridx1[r] = tglob; }
            }
        }
    }

    // ---- cross-lane min/argmin within each 16-lane half (wave32) ----
#pragma unroll
    for (int r = 0; r < 8; ++r) {
        float m0 = rmin0[r], m1 = rmin1[r];
        int   i0 = ridx0[r], i1 = ridx1[r];
#pragma unroll
        for (int s = 1; s <= 8; s <<= 1) {
            const float om0 = __shfl_xor(m0, s, 32);
            const int   oi0 = __shfl_xor(i0, s, 32);
            if (om0 < m0 || (om0 == m0 && oi0 < i0)) { m0 = om0; i0 = oi0; }
            const float om1 = __shfl_xor(m1, s, 32);
            const int   oi1 = __shfl_xor(i1, s, 32);
            if (om1 < m1 || (om1 == m1 && oi1 < i1)) { m1 = om1; i1 = oi1; }
        }
        rmin0[r] = m0; ridx0[r] = i0;
        rmin1[r] = m1; ridx1[r] = i1;
    }

    if (l16 == 0) {   // lane 0 -> rows q..q+7 ; lane 16 -> rows q+8..q+15
        const int qrow = qbase + (hi ? 8 : 0);
        float* __restrict__ om = outMin + (size_t)b * nQ;
        float* __restrict__ oi = outIdx + (size_t)b * nQ;
#pragma unroll
        for (int r = 0; r < 8; ++r) {
            om[qrow + r]      = fminf(fmaxf(rmin0[r], EPSV), DIST_MAXV);
            oi[qrow + r]      = (float)ridx0[r];
            om[qrow + 16 + r] = fminf(fmaxf(rmin1[r], EPSV), DIST_MAXV);
            oi[qrow + 16 + r] = (float)ridx1[r];
        }
    }
}

extern "C" void kernel_launch(void* const* d_in, const int* in_sizes, int n_in,
                              void* d_out, int out_size, void* d_ws, size_t ws_size,
                              hipStream_t stream) {
    (void)in_sizes; (void)n_in; (void)out_size; (void)d_ws; (void)ws_size;
    const int B = 4, N = 8192, M = 8192;

    const float* in1 = (const float*)d_in[0];   // [B, N, 3]
    const float* in2 = (const float*)d_in[1];   // [B, M, 3]
    float* out = (float*)d_out;

    float* min1 = out;                              // [B, N]
    float* min2 = out + (size_t)B * N;              // [B, M]
    float* idx1 = out + (size_t)B * N + (size_t)B * M;              // [B, N]
    float* idx2 = out + 2 * (size_t)B * N + (size_t)B * M;          // [B, M]

    dim3 block(256);
    dim3 grid1(N / 256, B);   // 8 waves/block * 32 queries/wave = 256 queries/block
    dim3 grid2(M / 256, B);

    // direction 1: x -> y
    chamfer_dir_kernel<<<grid1, block, 0, stream>>>(in1, in2, min1, idx1, N, M);
    // direction 2: y -> x
    chamfer_dir_kernel<<<grid2, block, 0, stream>>>(in2, in1, min2, idx2, M, N);
}